// FreqHash_28028956573735
// MI455X (gfx1250) — compile-verified
//
#include <hip/hip_runtime.h>
#include <cstdint>

#define NPB 64          // points per block
#define TPB 576         // threads per block (18 wave32)
#define GSTRIDE 516     // padded dwords per group slab (32*16 + 4 pad -> keeps 16B alignment)

__global__ __launch_bounds__(TPB, 1) void freqhash_kernel(
    const float* __restrict__ points,
    const float* __restrict__ features,
    float* __restrict__ out,
    int nPoints)
{
    // [g][i][c] layout, 4-dword pad per g: b128 taps stay 16B-aligned and
    // bank = (4g + 16i + 4c4) % 64 spreads the gather across banks.
    __shared__ __align__(16) float tableLDS[36 * GSTRIDE];  // 74,304 B
    __shared__ float4 weLDS[NPB * 36];                      // 36,864 B : {enc, w, rowIdx, -}

    const int t      = threadIdx.x;
    const int pBase  = blockIdx.x * NPB;
    const int pCount = min(NPB, nPoints - pBase);

    // ---- Stage feature table into LDS via CDNA5 async global->LDS DMA.
    //      Per-lane src/dst addressing performs the [g][c][i] -> [g][i][c]
    //      transpose + padding for free; tracked on ASYNCcnt and overlapped
    //      with the sincos phase below. ----
    const uint32_t ldsBase = (uint32_t)(uintptr_t)&tableLDS[0]; // low 32 bits of generic ptr = LDS byte offset
    for (int q = t; q < 36 * GSTRIDE; q += TPB) {
        int g   = q / GSTRIDE;
        int rem = q - g * GSTRIDE;
        int srcIdx;
        if (rem >= 512) {
            srcIdx = g * 512;                    // pad slot: harmless in-bounds load
        } else {
            int i = rem >> 4;                    // row     0..31
            int c = rem & 15;                    // channel 0..15
            srcIdx = g * 512 + c * 32 + i;       // features[g][c][i][0]
        }
        uint64_t gaddr = (uint64_t)(uintptr_t)(features + srcIdx);
        uint32_t laddr = ldsBase + (uint32_t)q * 4u;
        asm volatile("global_load_async_to_lds_b32 %0, %1, off"
                     :: "v"(laddr), "v"(gaddr) : "memory");
    }

    // ---- Phase A: per (point, freq, coord): one sincos feeds two groups.
    //      Precompute {enc, lerp weight, padded row index} once per (p,g). ----
    for (int k = t; k < pCount * 18; k += TPB) {
        int p = k / 18;
        int r = k - p * 18;
        int f = r / 3;
        int d = r - f * 3;
        float coord = points[(size_t)(pBase + p) * 3 + d];
        float freq  = (float)(1 << f);           // exactly 2**linspace(0,5,6)
        float sv, cv;
        sincosf(coord * freq, &sv, &cv);
        #pragma unroll
        for (int s = 0; s < 2; ++s) {
            float e  = s ? cv : sv;
            int   g  = f * 6 + s * 3 + d;        // (F,2,3) flattening order
            float iy = (e + 1.0f) * 15.5f;       // == ((e+1)*0.5)*31 (x0.5 is exact)
            iy = fminf(fmaxf(iy, 0.0f), 31.0f);
            int i0 = (int)iy;                    // iy >= 0 -> trunc == floor
            if (i0 > 30) i0 = 30;                // keeps taps in-slab; lerp identical
            float w  = iy - (float)i0;
            int   ri = g * GSTRIDE + i0 * 16;
            weLDS[p * 36 + g] = make_float4(e, w, __int_as_float(ri), 0.0f);
        }
    }

    asm volatile("s_wait_asynccnt 0x0" ::: "memory");   // DMA done before anyone reads table
    __syncthreads();

    // ---- Phase B: thread owns (pSub, c4, g): one b128 we-load + two b128
    //      table taps produce 4 channels -> 12 B LDS per 4 B output, and the
    //      4 stores (offsets 0/144/288/432 B) stay fully coalesced across the
    //      g-sweeping lanes. 604 MB of streaming stores is the roofline term. ----
    const int pSub = t / 144;          // 4 points in flight per block
    const int u    = t - pSub * 144;
    const int c4   = u / 36;           // channel quad 0..3
    const int g    = u - c4 * 36;      // group 0..35
    float* outP = out + (size_t)(pBase + pSub) * 576 + c4 * 144 + g;
    const float4* wePtr = &weLDS[pSub * 36 + g];
    for (int p = pSub; p < pCount; p += 4) {
        float4 we   = *wePtr;
        int    base = __float_as_int(we.z) + c4 * 4;
        float4 V0   = *(const float4*)&tableLDS[base];
        float4 V1   = *(const float4*)&tableLDS[base + 16];
        float  w    = we.y;
        float  omw  = 1.0f - w;
        float  e    = we.x;
        outP[0]   = fmaf(V1.x, w, V0.x * omw) + e;   // channel 4*c4+0
        outP[36]  = fmaf(V1.y, w, V0.y * omw) + e;   // channel 4*c4+1
        outP[72]  = fmaf(V1.z, w, V0.z * omw) + e;   // channel 4*c4+2
        outP[108] = fmaf(V1.w, w, V0.w * omw) + e;   // channel 4*c4+3
        wePtr += 4 * 36;
        outP  += 4 * 576;
    }
}

extern "C" void kernel_launch(void* const* d_in, const int* in_sizes, int n_in,
                              void* d_out, int out_size, void* d_ws, size_t ws_size,
                              hipStream_t stream) {
    const float* points   = (const float*)d_in[0];
    // d_in[1] = freqs: exact powers of two, recomputed in-kernel
    const float* features = (const float*)d_in[2];
    float* out = (float*)d_out;
    int n = in_sizes[0] / 3;
    int blocks = (n + NPB - 1) / NPB;
    freqhash_kernel<<<blocks, TPB, 0, stream>>>(points, features, out, n);
}